// GNNModel_47115791237137
// MI455X (gfx1250) — compile-verified
//
#include <hip/hip_runtime.h>
#include <math.h>

typedef __attribute__((ext_vector_type(2))) float v2f;
typedef __attribute__((ext_vector_type(8))) float v8f;

#define NNODES 50000
#define NEDGES 800000
#define DCH    64

// ---------------- degree / normalization ----------------

__global__ void deg_init_kernel(float* __restrict__ deg, int n) {
  int i = blockIdx.x * blockDim.x + threadIdx.x;
  if (i < n) deg[i] = 1.0f;   // self-loop contributes 1 to every node's degree
}

__global__ void deg_accum_kernel(const int* __restrict__ dst, float* __restrict__ deg, int e) {
  int i = blockIdx.x * blockDim.x + threadIdx.x;
  if (i < e) atomicAdd(&deg[dst[i]], 1.0f);
}

__global__ void dinv_kernel(float* __restrict__ deg, int n) {
  int i = blockIdx.x * blockDim.x + threadIdx.x;
  if (i < n) deg[i] = rsqrtf(deg[i]);   // deg >= 1 always (self-loop)
}

__global__ void coef_kernel(const int* __restrict__ src, const int* __restrict__ dst,
                            const float* __restrict__ dinv, float* __restrict__ coef, int e) {
  int i = blockIdx.x * blockDim.x + threadIdx.x;
  if (i < e) coef[i] = dinv[src[i]] * dinv[dst[i]];
}

// ---------------- GEMM: T = H @ W  (N x 64) @ (64 x 64), fp32 WMMA ----------------
// One wave computes a 16-row x 64-col tile. 256 threads = 8 waves = 128 rows/block.
// W staged in LDS with stride 65 (pad) to avoid bank conflicts between half-waves.

__global__ void __launch_bounds__(256)
gemm64_wmma_kernel(const float* __restrict__ H, const float* __restrict__ W,
                   float* __restrict__ T, int n) {
  __shared__ float sW[64 * 65];
  int tid = threadIdx.x;
  for (int idx = tid; idx < 64 * 64; idx += 256) {
    int r = idx >> 6, c = idx & 63;
    sW[r * 65 + c] = W[idx];
  }
  __syncthreads();

  int wave = tid >> 5;
  int lane = tid & 31;
  int rowTile = (blockIdx.x * 8 + wave) * 16;
  if (rowTile >= n) return;            // whole wave exits together; EXEC stays all-1s

  int m  = lane & 15;                  // row within A-tile / col within B-tile
  int kh = lane >> 4;                  // half-wave selector: K offset 0 or 2

  const float* hrow = H + (size_t)(rowTile + m) * DCH;

  v8f acc0 = {}, acc1 = {}, acc2 = {}, acc3 = {};

#pragma unroll
  for (int k = 0; k < 64; k += 4) {
    // A fragment (16x4 f32): lanes 0-15 -> K={k,k+1}, lanes 16-31 -> K={k+2,k+3}
    v2f a;
    a.x = hrow[k + 2 * kh];
    a.y = hrow[k + 2 * kh + 1];
    // B fragments (4x16 f32): VGPR0 = rows k / k+2, VGPR1 = rows k+1 / k+3
    const float* wr0 = &sW[(k + 2 * kh) * 65];
    const float* wr1 = wr0 + 65;
    {
      v2f b; b.x = wr0[0 * 16 + m]; b.y = wr1[0 * 16 + m];
      acc0 = __builtin_amdgcn_wmma_f32_16x16x4_f32(false, a, false, b, (short)0, acc0, false, false);
    }
    {
      v2f b; b.x = wr0[1 * 16 + m]; b.y = wr1[1 * 16 + m];
      acc1 = __builtin_amdgcn_wmma_f32_16x16x4_f32(false, a, false, b, (short)0, acc1, false, false);
    }
    {
      v2f b; b.x = wr0[2 * 16 + m]; b.y = wr1[2 * 16 + m];
      acc2 = __builtin_amdgcn_wmma_f32_16x16x4_f32(false, a, false, b, (short)0, acc2, false, false);
    }
    {
      v2f b; b.x = wr0[3 * 16 + m]; b.y = wr1[3 * 16 + m];
      acc3 = __builtin_amdgcn_wmma_f32_16x16x4_f32(false, a, false, b, (short)0, acc3, false, false);
    }
  }

  // C/D layout: VGPR j -> lanes 0-15: (M=j, N=lane), lanes 16-31: (M=j+8, N=lane-16)
  float* trow = T + (size_t)rowTile * DCH;
  int mbase = kh * 8;
#pragma unroll
  for (int j = 0; j < 8; ++j) {
    float* dst = trow + (size_t)(mbase + j) * DCH + m;
    dst[0]  = acc0[j];
    dst[16] = acc1[j];
    dst[32] = acc2[j];
    dst[48] = acc3[j];
  }
}

// ---------------- per-layer aggregation ----------------

// acc[i][c] = dinv[i]^2 * t[i][c] + b[c]   (self-loop folded in, no atomics needed)
__global__ void init_acc_kernel(const float* __restrict__ t, const float* __restrict__ dinv,
                                const float* __restrict__ b, float* __restrict__ acc, int n) {
  int idx = blockIdx.x * blockDim.x + threadIdx.x;
  if (idx >= n * DCH) return;
  int i = idx >> 6, c = idx & 63;
  float d = dinv[i];
  acc[idx] = d * d * t[idx] + b[c];
}

// acc[dst[e]][c] += coef[e] * t[src[e]][c]   -- 4 channels per thread
__global__ void scatter_kernel(const float* __restrict__ t, const float* __restrict__ coef,
                               const int* __restrict__ src, const int* __restrict__ dst,
                               float* __restrict__ acc, int e) {
  int idx = blockIdx.x * blockDim.x + threadIdx.x;
  if (idx >= e * 16) return;
  int i  = idx >> 4;
  int c4 = (idx & 15) << 2;
  int s = src[i];
  int d = dst[i];
  float cf = coef[i];
  const float4 tv = *(const float4*)(t + (size_t)s * DCH + c4);
  float* ap = acc + (size_t)d * DCH + c4;
  atomicAdd(ap + 0, cf * tv.x);
  atomicAdd(ap + 1, cf * tv.y);
  atomicAdd(ap + 2, cf * tv.z);
  atomicAdd(ap + 3, cf * tv.w);
}

__global__ void relu_kernel(float* __restrict__ a, int total) {
  int i = blockIdx.x * blockDim.x + threadIdx.x;
  if (i < total) a[i] = fmaxf(a[i], 0.0f);
}

// ---------------- driver ----------------

extern "C" void kernel_launch(void* const* d_in, const int* in_sizes, int n_in,
                              void* d_out, int out_size, void* d_ws, size_t ws_size,
                              hipStream_t stream) {
  const float* x  = (const float*)d_in[0];
  const int*   ei = (const int*)  d_in[1];   // [2, E] flat: src then dst
  const float* W1 = (const float*)d_in[2];
  const float* b1 = (const float*)d_in[3];
  const float* W2 = (const float*)d_in[4];
  const float* b2 = (const float*)d_in[5];
  const float* W3 = (const float*)d_in[6];
  const float* b3 = (const float*)d_in[7];

  const int* src = ei;
  const int* dst = ei + NEDGES;

  // carve workspace (256B-aligned chunks)
  char* ws = (char*)d_ws;
  size_t off = 0;
  auto carve = [&](size_t bytes) -> float* {
    float* p = (float*)(ws + off);
    off += (bytes + 255) & ~(size_t)255;
    return p;
  };
  float* dinv = carve((size_t)NNODES * 4);            // degree -> dinv (in place)
  float* coef = carve((size_t)NEDGES * 4);
  float* T    = carve((size_t)NNODES * DCH * 4);      // GEMM output
  float* A    = carve((size_t)NNODES * DCH * 4);      // aggregation acc == next h
  float* OUT  = (float*)d_out;

  const int TB = 256;
  int nElems   = NNODES * DCH;                        // 3.2M
  int gN       = (NNODES + TB - 1) / TB;
  int gE       = (NEDGES + TB - 1) / TB;
  int gElems   = (nElems + TB - 1) / TB;
  int gScatter = (NEDGES * 16 + TB - 1) / TB;
  int gGemm    = (NNODES / 16 + 7) / 8;               // 3125 row-tiles, 8 waves/block

  // normalization (shared across layers)
  deg_init_kernel<<<gN, TB, 0, stream>>>(dinv, NNODES);
  deg_accum_kernel<<<gE, TB, 0, stream>>>(dst, dinv, NEDGES);
  dinv_kernel<<<gN, TB, 0, stream>>>(dinv, NNODES);
  coef_kernel<<<gE, TB, 0, stream>>>(src, dst, dinv, coef, NEDGES);

  // layer 1: h = relu(Agg(x @ W1) + b1)
  gemm64_wmma_kernel<<<gGemm, TB, 0, stream>>>(x, W1, T, NNODES);
  init_acc_kernel<<<gElems, TB, 0, stream>>>(T, dinv, b1, A, NNODES);
  scatter_kernel<<<gScatter, TB, 0, stream>>>(T, coef, src, dst, A, NEDGES);
  relu_kernel<<<gElems, TB, 0, stream>>>(A, nElems);

  // layer 2: h = relu(Agg(h @ W2) + b2)   (A reused: gemm consumes it first)
  gemm64_wmma_kernel<<<gGemm, TB, 0, stream>>>(A, W2, T, NNODES);
  init_acc_kernel<<<gElems, TB, 0, stream>>>(T, dinv, b2, A, NNODES);
  scatter_kernel<<<gScatter, TB, 0, stream>>>(T, coef, src, dst, A, NEDGES);
  relu_kernel<<<gElems, TB, 0, stream>>>(A, nElems);

  // layer 3: out = Agg(h @ W3) + b3  (no relu), accumulate straight into d_out
  gemm64_wmma_kernel<<<gGemm, TB, 0, stream>>>(A, W3, T, NNODES);
  init_acc_kernel<<<gElems, TB, 0, stream>>>(T, dinv, b3, OUT, NNODES);
  scatter_kernel<<<gScatter, TB, 0, stream>>>(T, coef, src, dst, OUT, NEDGES);
}